// KalmanFilter_44289702756621
// MI455X (gfx1250) — compile-verified
//
#include <hip/hip_runtime.h>

// ---------------------------------------------------------------------------
// Kalman filter step on MI455X (gfx1250, wave32).
// All GEMMs (~47 GFLOP fp32) via V_WMMA_F32_16X16X4_F32; A/B tiles staged into
// LDS by the Tensor Data Mover (tensor_load_to_lds + s_wait_tensorcnt) where
// the tile is a plain row-major 2D copy; fragments are single ds_load_b64s.
// ---------------------------------------------------------------------------

typedef __attribute__((ext_vector_type(2))) float    v2f;
typedef __attribute__((ext_vector_type(8))) float    v8f;
typedef __attribute__((ext_vector_type(4))) unsigned u32x4;
typedef __attribute__((ext_vector_type(8))) int      i32x8;
typedef __attribute__((ext_vector_type(4))) int      i32x4;

static constexpr int kD = 1024;
static constexpr int kM = 512;
static constexpr int kC = 256;
static constexpr int kB = 8192;

static constexpr int LDS_STRIDE = 36;   // 32 data dwords + 4 pad dwords (144 B)

// ---------------------------------------------------------------------------
// TDM: DMA a [rows x 32-dword] row-major tile into LDS with +4 dword padding
// per row (matches LDS_STRIDE). 2D descriptor: groups 2/3 zero.
//   D# group0: count=1 | lds_addr | global_addr[56:0] | type=2
//   D# group1: data_size=4B, pad_enable, pad_interval=32dw(code 4),
//              pad_amount=4dw(code 3), tensor dims 1M (no clipping),
//              tile_dim0=32, tile_dim1=rows, stride0=row_stride
// This toolchain exposes the 6-arg builtin:
//   (u32x4 g0, i32x8 g1, i32x4 g2, i32x4 g3, i32x8 g4, i32 cpol)
// ---------------------------------------------------------------------------
__device__ __forceinline__ void tdm_load_tile(unsigned lds_addr, const float* gptr,
                                              int rows, int row_stride_dw)
{
    unsigned long long ga = (unsigned long long)(uintptr_t)gptr;
    u32x4 g0;
    g0.x = 0x1u;                                           // count=1, user mode
    g0.y = lds_addr;                                       // LDS byte address
    g0.z = (unsigned)(ga & 0xFFFFFFFFu);                   // global_addr[31:0]
    g0.w = (unsigned)((ga >> 32) & 0x01FFFFFFu) | (2u << 30); // [56:32] | type=2
    i32x8 g1;
    g1[0] = (int)((2u << 16) | (1u << 20) | (4u << 22) | (3u << 25));
    g1[1] = 0;                                             // no atomic barrier
    g1[2] = 0x10;                                          // tensor_dim0 = 1<<20
    g1[3] = 0x10 | (32 << 16);                             // tensor_dim1 hi | tile_dim0=32
    g1[4] = rows & 0xFFFF;                                 // tile_dim1 (tile_dim2=0)
    g1[5] = row_stride_dw;                                 // tensor_dim0_stride[31:0]
    g1[6] = 0;
    g1[7] = 0;
    i32x4 gz4 = {0, 0, 0, 0};
    i32x8 gz8 = {0, 0, 0, 0, 0, 0, 0, 0};
    __builtin_amdgcn_tensor_load_to_lds(g0, g1, gz4, gz4, gz8, 0);
}

// ---------------------------------------------------------------------------
// Generic fp32 WMMA GEMM:  Out[M,N] = epilogue( A[M,K] * B, Cin )
//   TRANSB == 0 : B is [K,N] row-major;  TRANSB == 1 : B[k][n] = G[n][k]
//   EPI: 0 -> AB, 1 -> Cin + AB, 2 -> Cin - AB
// Grid (N/128, M/128); 256 threads = 8 waves; wave tile 64x32 = 4x2 WMMA accs.
// LDS tiles stored fragment-major: As[m][k], Bs[n][k] (stride 36 dwords), so
// every WMMA fragment is one aligned ds_load_b64.
// ---------------------------------------------------------------------------
template <int TRANSB, int EPI>
__global__ __launch_bounds__(256) void wmma_gemm_f32(
    const float* __restrict__ A, int lda,
    const float* __restrict__ B, int ldb,
    const float* __restrict__ Cin, int ldc,
    float* __restrict__ Out, int ldo,
    int K)
{
    __shared__ float As[128][LDS_STRIDE];
    __shared__ float Bs[128][LDS_STRIDE];

    const int blockM = blockIdx.y * 128;
    const int blockN = blockIdx.x * 128;
    const int t    = threadIdx.x;
    const int lane = t & 31;
    const int wave = t >> 5;          // 0..7
    const int m0 = (wave >> 2) * 64;
    const int n0 = (wave & 3) * 32;
    const int h = lane >> 4;          // lane half (f32 frag layout)
    const int r = lane & 15;

    v8f acc[4][2] = {};

    for (int kc = 0; kc < K; kc += 32) {
        // ---- stage A tile [128 x 32] via Tensor Data Mover (wave 0 only) ----
        if (wave == 0) {
            tdm_load_tile((unsigned)(uintptr_t)&As[0][0],
                          A + (size_t)blockM * lda + kc, 128, lda);
            if constexpr (TRANSB == 1) {
                // B tile is also a plain row-major 2D copy: G[n][k]
                tdm_load_tile((unsigned)(uintptr_t)&Bs[0][0],
                              B + (size_t)blockN * ldb + kc, 128, ldb);
            }
            __builtin_amdgcn_s_wait_tensorcnt(0);
        }
        if constexpr (TRANSB == 0) {
            // ---- transpose-stage B [K-major -> n-major]: coalesced dword
            //      loads along n, contiguous b128 stores into Bs[n][k] ----
#pragma unroll
            for (int i = 0; i < 4; ++i) {
                int idx = t + i * 256;        // 0..1023
                int n   = idx & 127;
                int kq  = (idx >> 7) << 2;    // 0,4,...,28
                float4 v;
                v.x = B[(size_t)(kc + kq + 0) * ldb + blockN + n];
                v.y = B[(size_t)(kc + kq + 1) * ldb + blockN + n];
                v.z = B[(size_t)(kc + kq + 2) * ldb + blockN + n];
                v.w = B[(size_t)(kc + kq + 3) * ldb + blockN + n];
                *reinterpret_cast<float4*>(&Bs[n][kq]) = v;
            }
            if (kc + 32 < K)
                __builtin_prefetch(B + (size_t)(kc + 32 + (t >> 5)) * ldb + blockN + (lane << 2), 0, 1);
        }
        __syncthreads();

        // ---- 8 WMMA k-steps; frags are single aligned b64 LDS loads ----
#pragma unroll
        for (int kk = 0; kk < 32; kk += 4) {
            v2f a[4], b[2];
#pragma unroll
            for (int mi = 0; mi < 4; ++mi)
                a[mi] = *reinterpret_cast<const v2f*>(&As[m0 + mi * 16 + r][kk + 2 * h]);
#pragma unroll
            for (int ni = 0; ni < 2; ++ni)
                b[ni] = *reinterpret_cast<const v2f*>(&Bs[n0 + ni * 16 + r][kk + 2 * h]);
#pragma unroll
            for (int mi = 0; mi < 4; ++mi)
#pragma unroll
                for (int ni = 0; ni < 2; ++ni)
                    acc[mi][ni] = __builtin_amdgcn_wmma_f32_16x16x4_f32(
                        false, a[mi], false, b[ni],
                        (short)0, acc[mi][ni], false, false);
        }
        __syncthreads();
    }

    // ---- epilogue: C/D layout — VGPR i holds row (i + 8*h), col = lane&15 ----
#pragma unroll
    for (int mi = 0; mi < 4; ++mi)
#pragma unroll
        for (int ni = 0; ni < 2; ++ni)
#pragma unroll
            for (int i = 0; i < 8; ++i) {
                int row = blockM + m0 + mi * 16 + i + 8 * h;
                int col = blockN + n0 + ni * 16 + r;
                float v = acc[mi][ni][i];
                if constexpr (EPI == 1) v = Cin[(size_t)row * ldc + col] + v;
                if constexpr (EPI == 2) v = Cin[(size_t)row * ldc + col] - v;
                Out[(size_t)row * ldo + col] = v;
            }
}

// ---------------------------------------------------------------------------
// Gauss-Jordan inverse of 512x512 S (diagonally dominant; no pivot search).
// aug = [S | I] (512x1024); right half becomes S^-1. One 1024-thread block.
// ---------------------------------------------------------------------------
__global__ __launch_bounds__(1024) void invert_gauss_jordan(
    const float* __restrict__ S, float* __restrict__ aug)
{
    const int t = threadIdx.x;
    const int N = 512, W = 1024;
    __shared__ float piv[1024];
    __shared__ float fac[512];

    for (int idx = t; idx < N * W; idx += 1024) {
        int rr = idx >> 10, cc = idx & 1023;
        aug[idx] = (cc < N) ? S[rr * N + cc] : ((cc - N == rr) ? 1.0f : 0.0f);
    }
    __syncthreads();

    for (int k = 0; k < N; ++k) {
        float pval = aug[(size_t)k * W + k];
        float rowv = aug[(size_t)k * W + t];
        if (t < N) fac[t] = (t == k) ? 0.0f : aug[(size_t)t * W + k];
        __syncthreads();

        float v = rowv / pval;
        piv[t] = v;
        aug[(size_t)k * W + t] = v;
        __syncthreads();

        for (int idx = t; idx < N * W; idx += 1024) {
            int rr = idx >> 10, cc = idx & 1023;
            aug[idx] -= fac[rr] * piv[cc];
        }
        __syncthreads();
    }
}

// ---------------------------------------------------------------------------
// Launch sequence
// Inputs: state[D,B], state_cov[D,D], meas[M,B], control[C,B],
//         F[D,D], Q[D,D], Bc[D,C], H[M,D], R[M,M]
// Output: state_n[D,B] ++ cov_n[D,D]
// ---------------------------------------------------------------------------
extern "C" void kernel_launch(void* const* d_in, const int* in_sizes, int n_in,
                              void* d_out, int out_size, void* d_ws, size_t ws_size,
                              hipStream_t stream)
{
    const float* state = (const float*)d_in[0];
    const float* Pcov  = (const float*)d_in[1];
    const float* meas  = (const float*)d_in[2];
    const float* ctrl  = (const float*)d_in[3];
    const float* F     = (const float*)d_in[4];
    const float* Q     = (const float*)d_in[5];
    const float* Bc    = (const float*)d_in[6];
    const float* H     = (const float*)d_in[7];
    const float* R     = (const float*)d_in[8];

    float* state_out = (float*)d_out;                       // state_p lives here too
    float* cov_out   = (float*)d_out + (size_t)kD * kB;     // cov_p lives here too

    float* ws    = (float*)d_ws;
    float* tmpDD = ws;                          // [D,D]  F @ P
    float* innov = tmpDD + (size_t)kD * kD;     // [M,B]  meas - H @ state_p
    float* T2    = innov + (size_t)kM * kB;     // [M,D]  H @ cov_p
    float* Smat  = T2    + (size_t)kM * kD;     // [M,M]
    float* aug   = Smat  + (size_t)kM * kM;     // [M,2M] Gauss-Jordan
    float* PHt   = aug   + (size_t)kM * 2 * kM; // [D,M]
    float* Kg    = PHt   + (size_t)kD * kM;     // [D,M]

    const dim3 blk(256);

    // 1) state_p = Bc @ control
    wmma_gemm_f32<0, 0><<<dim3(kB / 128, kD / 128), blk, 0, stream>>>(
        Bc, kC, ctrl, kB, nullptr, 0, state_out, kB, kC);
    // 2) state_p = F @ state + state_p  (in place)
    wmma_gemm_f32<0, 1><<<dim3(kB / 128, kD / 128), blk, 0, stream>>>(
        F, kD, state, kB, state_out, kB, state_out, kB, kD);
    // 3) tmpDD = F @ P
    wmma_gemm_f32<0, 0><<<dim3(kD / 128, kD / 128), blk, 0, stream>>>(
        F, kD, Pcov, kD, nullptr, 0, tmpDD, kD, kD);
    // 4) cov_p = tmpDD @ F^T + Q
    wmma_gemm_f32<1, 1><<<dim3(kD / 128, kD / 128), blk, 0, stream>>>(
        tmpDD, kD, F, kD, Q, kD, cov_out, kD, kD);
    // 5) innov = meas - H @ state_p
    wmma_gemm_f32<0, 2><<<dim3(kB / 128, kM / 128), blk, 0, stream>>>(
        H, kD, state_out, kB, meas, kB, innov, kB, kD);
    // 6) T2 = H @ cov_p
    wmma_gemm_f32<0, 0><<<dim3(kD / 128, kM / 128), blk, 0, stream>>>(
        H, kD, cov_out, kD, nullptr, 0, T2, kD, kD);
    // 7) S = T2 @ H^T + R
    wmma_gemm_f32<1, 1><<<dim3(kM / 128, kM / 128), blk, 0, stream>>>(
        T2, kD, H, kD, R, kM, Smat, kM, kD);
    // 8) S^-1 into right half of aug
    invert_gauss_jordan<<<1, 1024, 0, stream>>>(Smat, aug);
    // 9) PHt = cov_p @ H^T
    wmma_gemm_f32<1, 0><<<dim3(kM / 128, kD / 128), blk, 0, stream>>>(
        cov_out, kD, H, kD, nullptr, 0, PHt, kM, kD);
    // 10) Kg = PHt @ S^-1   (S^-1 = aug right half, ld = 1024)
    wmma_gemm_f32<0, 0><<<dim3(kM / 128, kD / 128), blk, 0, stream>>>(
        PHt, kM, aug + kM, 2 * kM, nullptr, 0, Kg, kM, kM);
    // 11) state_n = state_p + Kg @ innov  (in place)
    wmma_gemm_f32<0, 1><<<dim3(kB / 128, kD / 128), blk, 0, stream>>>(
        Kg, kM, innov, kB, state_out, kB, state_out, kB, kM);
    // 12) cov_n = cov_p - Kg @ T2        (in place)
    wmma_gemm_f32<0, 2><<<dim3(kD / 128, kD / 128), blk, 0, stream>>>(
        Kg, kM, T2, kD, cov_out, kD, cov_out, kD, kM);

    (void)in_sizes; (void)n_in; (void)out_size; (void)ws_size;
}